// DerivativeNet_32332513804927
// MI455X (gfx1250) — compile-verified
//
#include <hip/hip_runtime.h>

#define BB 16
#define CC 2
#define HH 1024
#define WW 1024

// gfx1250 async global->LDS path (probe-confirmed to compile in round 2).
#if defined(__AMDGCN__) && __has_builtin(__builtin_amdgcn_global_load_async_to_lds_b128) && __has_builtin(__builtin_amdgcn_s_wait_asynccnt)
#define USE_ASYNC_LDS 1
#else
#define USE_ASYNC_LDS 0
#endif

#if USE_ASYNC_LDS
typedef int v4i __attribute__((ext_vector_type(4)));
typedef __attribute__((address_space(1))) v4i* gv4i_p;   // global
typedef __attribute__((address_space(3))) v4i* lv4i_p;   // LDS
#endif

__global__ __launch_bounds__(256) void deriv_row_kernel(const float* __restrict__ u,
                                                        const float* __restrict__ mask,
                                                        float* __restrict__ out) {
  __shared__ float    u_lds[CC][WW];     // staged u rows (both channels)
  __shared__ unsigned bits[WW / 32];     // 1024-bit mask bitmap for this row
  __shared__ int      sFirst, sSecond, sLast;

  const int tid = threadIdx.x;           // 0..255, each owns columns 4t..4t+3
  const int row = blockIdx.x;            // 0..B*H-1
  const int b   = row >> 10;
  const int h   = row & (HH - 1);

  const size_t mrow  = ((size_t)b * HH + h) * WW;
  const size_t urow0 = (((size_t)b * CC + 0) * HH + h) * WW;
  const size_t urow1 = (((size_t)b * CC + 1) * HH + h) * WW;

  // ---- stage u rows (both channels) into LDS, 16B per thread per channel ----
#if USE_ASYNC_LDS
  {
    float* g0 = const_cast<float*>(u) + urow0 + 4 * tid;
    float* g1 = const_cast<float*>(u) + urow1 + 4 * tid;
    __builtin_amdgcn_global_load_async_to_lds_b128(
        (gv4i_p)g0, (lv4i_p)(&u_lds[0][4 * tid]), 0, 0);
    __builtin_amdgcn_global_load_async_to_lds_b128(
        (gv4i_p)g1, (lv4i_p)(&u_lds[1][4 * tid]), 0, 0);
  }
#else
  *(float4*)(&u_lds[0][4 * tid]) = *(const float4*)(u + urow0 + 4 * tid);
  *(float4*)(&u_lds[1][4 * tid]) = *(const float4*)(u + urow1 + 4 * tid);
#endif

  // ---- build the row's mask bitmap: shfl-OR nibbles within 8-lane groups ----
  const float4 m4 = *(const float4*)(mask + mrow + 4 * tid);
  unsigned nib = (m4.x != 0.0f ? 1u : 0u) | (m4.y != 0.0f ? 2u : 0u) |
                 (m4.z != 0.0f ? 4u : 0u) | (m4.w != 0.0f ? 8u : 0u);
  unsigned v = nib << ((tid & 7) * 4);
  v |= __shfl_xor(v, 1);
  v |= __shfl_xor(v, 2);
  v |= __shfl_xor(v, 4);
  if ((tid & 7) == 0) bits[tid >> 3] = v;   // word (tid>>3) = columns 32*(tid>>3)..+31
  __syncthreads();

  // ---- wave-parallel scan: first / second / last set bit of the row ----
  if (tid < 32) {                          // exactly wave 0, all 32 lanes
    const unsigned bv = bits[tid];
    const unsigned wm = (unsigned)__ballot(bv != 0u);  // nonzero-word mask (uniform)
    int f = 1 << 30, s = 1 << 30, l = -1;
    if (wm) {
      const int w0 = __builtin_ctz(wm);
      const unsigned b0 = bits[w0];
      f = w0 * 32 + __builtin_ctz(b0);
      const unsigned r = b0 & (b0 - 1u);   // clear lowest bit
      if (r) {
        s = w0 * 32 + __builtin_ctz(r);
      } else {
        const unsigned wm2 = wm & (wm - 1u);
        if (wm2) {
          const int w1 = __builtin_ctz(wm2);
          s = w1 * 32 + __builtin_ctz(bits[w1]);
        } else {
          s = WW;                          // single masked pixel: edge1 to row end
        }
      }
      const int wl = 31 - __builtin_clz(wm);
      l = wl * 32 + (31 - __builtin_clz(bits[wl]));
    }
    if (tid == 0) { sFirst = f; sSecond = s; sLast = l; }
  }

#if USE_ASYNC_LDS
  __builtin_amdgcn_s_wait_asynccnt(0);     // our async copies landed in LDS
#endif
  __syncthreads();                          // bitmap + scalars + u_lds visible

  const int   first = sFirst, second = sSecond, last = sLast;
  const float inv2h = 50.0f;               // 1/(2*0.01)
  const float invh  = 100.0f;              // 1/0.01

  float a0[4], a1[4];
  for (int k = 0; k < 4; ++k) {
    const int w = 4 * tid + k;
    const unsigned mbm1 = (w > 0)      ? ((bits[(w - 1) >> 5] >> ((w - 1) & 31)) & 1u) : 0u;
    const unsigned mb   =                ((bits[w >> 5] >> (w & 31)) & 1u);
    const unsigned mbp1 = (w < WW - 1) ? ((bits[(w + 1) >> 5] >> ((w + 1) & 31)) & 1u) : 0u;

    const float er = (float)(mbm1 & mb & mbp1);
    const float e1 = (w >= first && w < second) ? 1.0f : 0.0f;
    const float e2 = (w == last) ? 1.0f : 0.0f;

    {
      const float um1 = (w > 0)      ? u_lds[0][w - 1] : 0.0f;
      const float u0  =                u_lds[0][w];
      const float up1 = (w < WW - 1) ? u_lds[0][w + 1] : 0.0f;
      a0[k] = er * (up1 - um1) * inv2h + e1 * (up1 - u0) * invh + e2 * (u0 - um1) * invh;
    }
    {
      const float um1 = (w > 0)      ? u_lds[1][w - 1] : 0.0f;
      const float u0  =                u_lds[1][w];
      const float up1 = (w < WW - 1) ? u_lds[1][w + 1] : 0.0f;
      a1[k] = er * (up1 - um1) * inv2h + e1 * (up1 - u0) * invh + e2 * (u0 - um1) * invh;
    }
  }

  *(float4*)(out + urow0 + 4 * tid) = make_float4(a0[0], a0[1], a0[2], a0[3]);
  *(float4*)(out + urow1 + 4 * tid) = make_float4(a1[0], a1[1], a1[2], a1[3]);
}

extern "C" void kernel_launch(void* const* d_in, const int* in_sizes, int n_in,
                              void* d_out, int out_size, void* d_ws, size_t ws_size,
                              hipStream_t stream) {
  const float* u    = (const float*)d_in[0];
  const float* mask = (const float*)d_in[1];
  float*       out  = (float*)d_out;

  dim3 grid(BB * HH);   // one workgroup per (b,h) row: 16384 blocks
  dim3 block(256);      // 8 wave32 waves; 4 columns per thread
  hipLaunchKernelGGL(deriv_row_kernel, grid, block, 0, stream, u, mask, out);

  (void)in_sizes; (void)n_in; (void)out_size; (void)d_ws; (void)ws_size;
}